// net_62972810494302
// MI455X (gfx1250) — compile-verified
//
#include <hip/hip_runtime.h>
#include <stdint.h>

// ---------------------------------------------------------------------------
// Problem constants (reference: N=2,000,000 anchors, C=8 classes, 6 deltas)
// ---------------------------------------------------------------------------
#define NCLS   8
#define NDEL   6
#define TPB    256
#define ROUNDS 8
#define CHUNK  (TPB * ROUNDS)   // 2048 anchors per block -> 977 blocks for 2M

#define AS_GLOBAL __attribute__((address_space(1)))
#define AS_LOCAL  __attribute__((address_space(3)))

typedef int v4i __attribute__((ext_vector_type(4)));

// Async-to-LDS path: probe via __has_builtin (device pass only; host pass
// parses the fallback body).
#if defined(__HIP_DEVICE_COMPILE__) && __has_builtin(__builtin_amdgcn_global_load_async_to_lds_b128)
#define HAVE_ASYNC 1
#else
#define HAVE_ASYNC 0
#endif

#if HAVE_ASYNC
__device__ __forceinline__ void async_copy16(const float* g, float* lds) {
  // Builtin signature (from clang diagnostic): param0 is `int4 AS(1)*`.
  __builtin_amdgcn_global_load_async_to_lds_b128(
      (AS_GLOBAL v4i*)(g), (AS_LOCAL v4i*)(lds), 0, 0);
}
__device__ __forceinline__ void wait_async_le2() {
#if __has_builtin(__builtin_amdgcn_s_wait_asynccnt)
  __builtin_amdgcn_s_wait_asynccnt(2);
#else
  asm volatile("s_wait_asynccnt 0x2" ::: "memory");
#endif
}
__device__ __forceinline__ void wait_async_le0() {
#if __has_builtin(__builtin_amdgcn_s_wait_asynccnt)
  __builtin_amdgcn_s_wait_asynccnt(0);
#else
  asm volatile("s_wait_asynccnt 0x0" ::: "memory");
#endif
}
#endif

// ---------------------------------------------------------------------------
// wave32 shuffle helpers
// ---------------------------------------------------------------------------
__device__ __forceinline__ float waveRedF(float v) {
#pragma unroll
  for (int o = 16; o > 0; o >>= 1) v += __shfl_down(v, o, 32);
  return v;
}
__device__ __forceinline__ int waveRedI(int v) {
#pragma unroll
  for (int o = 16; o > 0; o >>= 1) v += __shfl_down(v, o, 32);
  return v;
}
__device__ __forceinline__ int waveScanIncl(int v, int lane) {
#pragma unroll
  for (int o = 1; o < 32; o <<= 1) {
    int u = __shfl_up(v, o, 32);
    if (lane >= o) v += u;
  }
  return v;
}

// ---------------------------------------------------------------------------
// Focal math for one anchor (8 logits in registers; dynamic class index
// resolved with a cndmask chain -> no scratch spill)
// ---------------------------------------------------------------------------
__device__ __forceinline__ void focalAccum(const float l[NCLS], int m,
                                           float& facc, int& nn, int& np) {
  int tgt = m > 0 ? m : 0;
  float mx = l[0];
#pragma unroll
  for (int j = 1; j < NCLS; ++j) mx = fmaxf(mx, l[j]);
  float se = 0.f;
#pragma unroll
  for (int j = 0; j < NCLS; ++j) se += __expf(l[j] - mx);
  float lt = l[0];
#pragma unroll
  for (int j = 1; j < NCLS; ++j) lt = (tgt == j) ? l[j] : lt;
  float logq = lt - mx - __logf(se);
  float q    = __expf(logq);
  float om   = 1.f - q;
  float fl   = -(om * om) * logq;   // gamma == 2
  if (m != 0) { facc += fl; nn += 1; }
  if (m > 0)  np += 1;
}

// ---------------------------------------------------------------------------
// Kernel A: per-block focal sum, non-neutral count, positive count
// ---------------------------------------------------------------------------
__global__ __launch_bounds__(TPB) void k_focal(
    const float* __restrict__ logits, const int* __restrict__ matches, int n,
    float* __restrict__ blkFocal, int* __restrict__ blkNN,
    int* __restrict__ blkPos) {
  const int tid = threadIdx.x;
  const int blk = blockIdx.x;
  const long long base = (long long)blk * CHUNK;
  const int lane = tid & 31, wid = tid >> 5;

  float facc = 0.f;
  int nn = 0, np = 0;

#if HAVE_ASYNC
  // Double-buffered async staging: each lane copies its own 32B logits row
  // into LDS and later consumes it itself -> only per-wave asynccnt waits.
  __shared__ __align__(16) float lbuf[2][TPB * NCLS];
  {
    long long idx = base + tid;
    long long src = idx < n ? idx : (long long)(n - 1);  // clamp: keep EXEC full
    const float* g = logits + src * NCLS;
    async_copy16(g,     &lbuf[0][tid * NCLS]);
    async_copy16(g + 4, &lbuf[0][tid * NCLS + 4]);
  }
#pragma unroll
  for (int r = 0; r < ROUNDS; ++r) {
    if (r + 1 < ROUNDS) {
      long long idx = base + (long long)(r + 1) * TPB + tid;
      long long src = idx < n ? idx : (long long)(n - 1);
      const float* g = logits + src * NCLS;
      const int b = (r + 1) & 1;
      async_copy16(g,     &lbuf[b][tid * NCLS]);
      async_copy16(g + 4, &lbuf[b][tid * NCLS + 4]);
      wait_async_le2();   // next tile (2 ops) may remain in flight
    } else {
      wait_async_le0();
    }
    long long idx = base + (long long)r * TPB + tid;
    if (idx < n) {
      const float4* lp = (const float4*)&lbuf[r & 1][tid * NCLS];
      float4 a = lp[0], b4 = lp[1];
      float l[NCLS] = {a.x, a.y, a.z, a.w, b4.x, b4.y, b4.z, b4.w};
      int m = matches[idx];
      focalAccum(l, m, facc, nn, np);
    }
  }
#else
#pragma unroll
  for (int r = 0; r < ROUNDS; ++r) {
    long long idx = base + (long long)r * TPB + tid;
    if (idx < n) {
      if (idx + TPB < n)   // CDNA5 global_prefetch_b8 for the next tile
        __builtin_prefetch(logits + (idx + TPB) * NCLS, 0, 0);
      const float4* gp = (const float4*)(logits + idx * NCLS);
      float4 a = gp[0], b4 = gp[1];
      float l[NCLS] = {a.x, a.y, a.z, a.w, b4.x, b4.y, b4.z, b4.w};
      int m = matches[idx];
      focalAccum(l, m, facc, nn, np);
    }
  }
#endif

  // deterministic block reduction
  __shared__ float sf[TPB / 32];
  __shared__ int   si[TPB / 32], sj[TPB / 32];
  facc = waveRedF(facc); nn = waveRedI(nn); np = waveRedI(np);
  if (lane == 0) { sf[wid] = facc; si[wid] = nn; sj[wid] = np; }
  __syncthreads();
  if (wid == 0) {
    float f = (lane < TPB / 32) ? sf[lane] : 0.f;
    int   a = (lane < TPB / 32) ? si[lane] : 0;
    int   b = (lane < TPB / 32) ? sj[lane] : 0;
    f = waveRedF(f); a = waveRedI(a); b = waveRedI(b);
    if (lane == 0) { blkFocal[blk] = f; blkNN[blk] = a; blkPos[blk] = b; }
  }
}

// ---------------------------------------------------------------------------
// Kernel B: exclusive scan of per-block positive counts (nb <= 1024)
// ---------------------------------------------------------------------------
__global__ __launch_bounds__(1024) void k_scan(const int* __restrict__ cnt,
                                               int* __restrict__ off, int nb) {
  __shared__ int wsum[32];
  const int t = threadIdx.x, lane = t & 31, wid = t >> 5;
  int v    = (t < nb) ? cnt[t] : 0;
  int incl = waveScanIncl(v, lane);
  if (lane == 31) wsum[wid] = incl;
  __syncthreads();
  if (wid == 0) {
    int w = wsum[lane];
    int s = waveScanIncl(w, lane);
    wsum[lane] = s - w;  // exclusive wave offsets
  }
  __syncthreads();
  if (t < nb) off[t] = wsum[wid] + incl - v;
}

// ---------------------------------------------------------------------------
// Kernel C: smooth-L1 with packed-target gather via dense positive ranks
// ---------------------------------------------------------------------------
__global__ __launch_bounds__(TPB) void k_bbox(
    const float* __restrict__ pred, const float* __restrict__ tgt,
    const int* __restrict__ matches, int n,
    const int* __restrict__ blkOff, float* __restrict__ blkSL1) {
  const int tid = threadIdx.x, blk = blockIdx.x;
  const long long base = (long long)blk * CHUNK;
  const int lane = tid & 31, wid = tid >> 5;
  __shared__ int wsum[TPB / 32];
  __shared__ int stot;

  int running = blkOff[blk];
  float acc = 0.f;

  for (int r = 0; r < ROUNDS; ++r) {
    long long idx = base + (long long)r * TPB + tid;
    int m = (idx < n) ? matches[idx] : 0;
    if (idx + TPB < n) {  // prefetch next tile's pred rows (global_prefetch_b8)
      __builtin_prefetch(pred + (idx + TPB) * NDEL, 0, 0);
      __builtin_prefetch(matches + idx + TPB, 0, 0);
    }
    int p = (m > 0) ? 1 : 0;

    __syncthreads();  // protect wsum/stot reuse across rounds
    int incl = waveScanIncl(p, lane);
    if (lane == 31) wsum[wid] = incl;
    __syncthreads();
    if (wid == 0) {
      int v = (lane < TPB / 32) ? wsum[lane] : 0;
      int s = waveScanIncl(v, lane);
      if (lane < TPB / 32) wsum[lane] = s - v;
      if (lane == TPB / 32 - 1) stot = s;
    }
    __syncthreads();
    int excl = wsum[wid] + incl - p;

    if (p) {
      long long rank = (long long)(running + excl);  // < n_pos <= n
      const float2* pp = (const float2*)(pred + idx * NDEL);   // 8B aligned
      const float2* tp = (const float2*)(tgt + rank * NDEL);   // 8B aligned
#pragma unroll
      for (int j = 0; j < 3; ++j) {
        float2 pv = pp[j], tv = tp[j];
        float d0 = fabsf(pv.x - tv.x);
        float d1 = fabsf(pv.y - tv.y);
        acc += (d0 < 1.f) ? 0.5f * d0 * d0 : d0 - 0.5f;
        acc += (d1 < 1.f) ? 0.5f * d1 * d1 : d1 - 0.5f;
      }
    }
    running += stot;
  }

  __shared__ float sf[TPB / 32];
  acc = waveRedF(acc);
  if (lane == 0) sf[wid] = acc;
  __syncthreads();
  if (wid == 0) {
    float f = (lane < TPB / 32) ? sf[lane] : 0.f;
    f = waveRedF(f);
    if (lane == 0) blkSL1[blk] = f;
  }
}

// ---------------------------------------------------------------------------
// Kernel D: final deterministic reductions + denominators
// ---------------------------------------------------------------------------
__global__ __launch_bounds__(1024) void k_final(
    const float* __restrict__ blkFocal, const int* __restrict__ blkNN,
    const float* __restrict__ blkSL1, const int* __restrict__ blkOff,
    const int* __restrict__ blkPos, int nb, float* __restrict__ out) {
  __shared__ float sf[32], sg[32];
  __shared__ int   si[32];
  const int t = threadIdx.x, lane = t & 31, wid = t >> 5;
  float f = (t < nb) ? blkFocal[t] : 0.f;
  int   a = (t < nb) ? blkNN[t]    : 0;
  float g = (t < nb) ? blkSL1[t]   : 0.f;
  f = waveRedF(f); a = waveRedI(a); g = waveRedF(g);
  if (lane == 0) { sf[wid] = f; si[wid] = a; sg[wid] = g; }
  __syncthreads();
  if (t == 0) {
    float F = 0.f, G = 0.f; int A = 0;
    for (int i = 0; i < 32; ++i) { F += sf[i]; A += si[i]; G += sg[i]; }
    int npos = blkOff[nb - 1] + blkPos[nb - 1];
    float cls = F / fmaxf((float)A, 1.f);
    float bb  = (npos > 0) ? G / fmaxf((float)npos * (float)NDEL, 1.f) : 0.f;
    out[0] = cls;
    out[1] = bb;
  }
}

// ---------------------------------------------------------------------------
// Host launcher
// ---------------------------------------------------------------------------
extern "C" void kernel_launch(void* const* d_in, const int* in_sizes, int n_in,
                              void* d_out, int out_size, void* d_ws,
                              size_t ws_size, hipStream_t stream) {
  (void)n_in; (void)out_size; (void)ws_size;
  const float* logits  = (const float*)d_in[0];
  const float* pred    = (const float*)d_in[1];
  const float* tgt     = (const float*)d_in[2];
  const int*   matches = (const int*)d_in[3];
  const int n  = in_sizes[3];
  const int nb = (n + CHUNK - 1) / CHUNK;   // 977 for N=2M (must be <= 1024)

  float* blkFocal = (float*)d_ws;
  int*   blkNN    = (int*)(blkFocal + nb);
  int*   blkPos   = blkNN + nb;
  int*   blkOff   = blkPos + nb;
  float* blkSL1   = (float*)(blkOff + nb);
  float* out      = (float*)d_out;

  k_focal<<<nb, TPB, 0, stream>>>(logits, matches, n, blkFocal, blkNN, blkPos);
  k_scan<<<1, 1024, 0, stream>>>(blkPos, blkOff, nb);
  k_bbox<<<nb, TPB, 0, stream>>>(pred, tgt, matches, n, blkOff, blkSL1);
  k_final<<<1, 1024, 0, stream>>>(blkFocal, blkNN, blkSL1, blkOff, blkPos, nb,
                                  out);
}